// ChEBIRecNN_43868795961379
// MI455X (gfx1250) — compile-verified
//
#include <hip/hip_runtime.h>
#include <math.h>

// ---------------- problem constants ----------------
#define D_   256
#define L_   64
#define K_   64
#define P_   8
#define F_   104
#define C_   500
#define NP   128      // F padded to 8 N-tiles of 16 (uniform 4 tiles per wave)
#define WS   136      // weight LDS row stride: 2*WS mod 64 = 16 -> B loads conflict-free
#define TWOF 208      // 2*F, K-dim of the scan matmul (exact multiple of 4)
#define XS   209      // x row stride (odd -> conflict-free A column reads)

typedef __attribute__((ext_vector_type(2))) float v2f;
typedef __attribute__((ext_vector_type(8))) float v8f;

// ---------------- LDS layout (in floats) ----------------
#define OFF_PREV 0                      // prev state: 64 x 104
#define OFF_X    (OFF_PREV + K_*F_)     // concat buffer: 64 x XS
#define OFF_W    (OFF_X + K_*XS)        // weights: 208 x WS (W1 first, then Wm)
#define OFF_IDX  (OFF_W + TWOF*WS)      // 512 ints
#define OFF_ATT  (OFF_IDX + K_*P_)
#define OFF_BM   (OFF_ATT + F_)
#define OFF_B1   (OFF_BM + F_)
#define LDS_FLOATS (OFF_B1 + F_)

// out[64 x 104] = relu( x[64 x kdim] @ w[kdim x 128pad] + bias ), fp32 WMMA.
//
// Tile grid: 4 M-tiles x 8 N-tiles (cols 104..127 zero-padded). Every wave
// owns ONE M-tile and exactly FOUR N-tiles: mt = wave&3, nt = (wave>>2)+2j.
// kb-outer / j-inner: one A fragment feeds 4 WMMAs on independent
// accumulators -> ~1.25 ds_loads per WMMA and no serial accumulate chain.
// No conditionals in the compute loop and wave id forced scalar via
// readfirstlane, so EXEC stays all-ones at every v_wmma (ISA 7.12 req).
__device__ __forceinline__ void mm_relu(const float* __restrict__ x,
                                        const float* __restrict__ w,
                                        float* __restrict__ out,
                                        const float* __restrict__ bias,
                                        int kdim)
{
    const int lane = threadIdx.x & 31;
    const int wave = __builtin_amdgcn_readfirstlane(threadIdx.x >> 5); // scalar
    const int half = lane >> 4;    // 0: lanes 0-15, 1: lanes 16-31
    const int l16  = lane & 15;

    const int mt  = wave & 3;      // M tile (scalar, constant per wave)
    const int nt0 = wave >> 2;     // 0 -> even N tiles, 1 -> odd N tiles (scalar)

    v8f acc[4] = {};
    // A: m = lane&15, k = kb + 2*half (+1 for second VGPR)
    const float* arow  = x + (mt * 16 + l16) * XS + 2 * half;
    // B: n = lane&15 within tile, k rows striped like A
    const float* bbase = w + nt0 * 16 + l16;

    for (int kb = 0; kb < kdim; kb += 4) {
        v2f a;
        a.x = arow[kb];
        a.y = arow[kb + 1];
        const float* brow = bbase + (kb + 2 * half) * WS;
#pragma unroll
        for (int j = 0; j < 4; ++j) {
            v2f b;
            b.x = brow[32 * j];
            b.y = brow[32 * j + WS];
            acc[j] = __builtin_amdgcn_wmma_f32_16x16x4_f32(
                false, a, false, b, (short)0, acc[j], false, false);
        }
    }

    // C/D 16x16 f32 layout: VGPR v -> row v + 8*half, col lane&15
#pragma unroll
    for (int j = 0; j < 4; ++j) {
        const int n = (nt0 + 2 * j) * 16 + l16;
        if (n < F_) {                      // plain store guard, no WMMA inside
            const float bn = bias[n];
#pragma unroll
            for (int v = 0; v < 8; ++v) {
                const int m = mt * 16 + v + 8 * half;
                out[m * F_ + n] = fmaxf(acc[j][v] + bn, 0.0f);
            }
        }
    }
}

__global__ void __launch_bounds__(256)
chain_kernel(const float* __restrict__ af, const int* __restrict__ pidx,
             const float* __restrict__ W1, const float* __restrict__ b1,
             const float* __restrict__ Wm, const float* __restrict__ bm,
             const float* __restrict__ att_w, float* __restrict__ sink_ws)
{
    extern __shared__ float smem[];
    float* prev = smem + OFF_PREV;
    float* x    = smem + OFF_X;
    float* w    = smem + OFF_W;
    int*   idx  = (int*)(smem + OFF_IDX);
    float* attw = smem + OFF_ATT;
    float* bmL  = smem + OFF_BM;
    float* b1L  = smem + OFF_B1;

    const int tid = threadIdx.x;
    const int d   = blockIdx.x;

    if (tid < F_) { attw[tid] = att_w[tid]; bmL[tid] = bm[tid]; b1L[tid] = b1[tid]; }

    // ---- t = 0: out0 = relu(atoms0 @ W1 + b1) ----
    for (int i = tid; i < F_ * NP; i += 256) {             // W1 -> w (N-padded)
        int r = i / NP, c = i - r * NP;
        w[r * WS + c] = (c < F_) ? W1[r * F_ + c] : 0.0f;
    }
    const float* a0 = af + (size_t)d * L_ * K_ * F_;
    for (int i = tid; i < K_ * F_; i += 256) {             // atoms0 -> x[:, 0:F)
        int k = i / F_, f = i - k * F_;
        x[k * XS + f] = a0[i];
    }
    __syncthreads();
    mm_relu(x, w, prev, b1L, F_);
    __syncthreads();
    for (int i = tid; i < TWOF * NP; i += 256) {           // Wm -> w (N-padded)
        int r = i / NP, c = i - r * NP;
        w[r * WS + c] = (c < F_) ? Wm[r * F_ + c] : 0.0f;
    }
    __syncthreads();

    // ---- scan t = 1..63 ----
    for (int t = 1; t < L_; ++t) {
        const int*   ip = pidx + ((size_t)d * (L_ - 1) + (t - 1)) * K_ * P_;
        const float* at = af + ((size_t)d * L_ + t) * (size_t)(K_ * F_);
        for (int i = tid; i < K_ * P_; i += 256) idx[i] = ip[i];
        for (int i = tid; i < K_ * F_; i += 256) {
            int k = i / F_, f = i - k * F_;
            x[k * XS + F_ + f] = at[i];                    // atoms -> x[:, F:2F)
            if (t + 1 < L_) __builtin_prefetch(at + K_ * F_ + i, 0, 0);
        }
        __syncthreads();

        // gather preds from prev via idx, softmax over P, agg -> x[:, 0:F)
        for (int i = tid; i < K_ * F_; i += 256) {
            int k = i / F_, f = i - k * F_;
            const float aw = attw[f];
            float vals[P_];
            float mx = -3.402823466e+38f;
#pragma unroll
            for (int p = 0; p < P_; ++p) {
                const int pi = idx[k * P_ + p];
                const float v = prev[pi * F_ + f];
                vals[p] = v;
                mx = fmaxf(mx, aw * v);
            }
            float se = 0.0f, sv = 0.0f;
#pragma unroll
            for (int p = 0; p < P_; ++p) {
                const float e = __expf(aw * vals[p] - mx);
                se += e;
                sv += e * vals[p];
            }
            x[k * XS + f] = sv / se;
        }
        __syncthreads();   // x complete; all reads of old prev done

        mm_relu(x, w, prev, bmL, TWOF);   // prev <- new state
        __syncthreads();
    }

    // sink = last[:, K-1]  (row k = 63)
    if (tid < F_) sink_ws[d * F_ + tid] = prev[63 * F_ + tid];
}

__global__ void __launch_bounds__(512)
finalize_kernel(const float* __restrict__ sink, const float* __restrict__ dag_w,
                const float* __restrict__ Wf, const float* __restrict__ bf,
                float* __restrict__ out)
{
    __shared__ float pooled[F_];
    const int tid = threadIdx.x;
    if (tid < F_) {
        const float w = dag_w[tid];
        float mx = -3.402823466e+38f;
        for (int d = 0; d < D_; ++d) mx = fmaxf(mx, w * sink[d * F_ + tid]);
        float se = 0.0f, sv = 0.0f;
        for (int d = 0; d < D_; ++d) {
            const float v = sink[d * F_ + tid];
            const float e = __expf(w * v - mx);
            se += e;
            sv += e * v;
        }
        pooled[tid] = sv / se;
    }
    __syncthreads();
    for (int c = tid; c < C_; c += 512) {
        float acc = bf[c];
        for (int f = 0; f < F_; ++f) acc += pooled[f] * Wf[f * C_ + c];
        out[c] = acc;
    }
}

extern "C" void kernel_launch(void* const* d_in, const int* in_sizes, int n_in,
                              void* d_out, int out_size, void* d_ws, size_t ws_size,
                              hipStream_t stream)
{
    const float* af   = (const float*)d_in[0];
    const int*   pidx = (const int*)d_in[1];
    const float* W1   = (const float*)d_in[2];
    const float* b1   = (const float*)d_in[3];
    const float* Wm   = (const float*)d_in[4];
    const float* bm   = (const float*)d_in[5];
    const float* attw = (const float*)d_in[6];
    const float* dagw = (const float*)d_in[7];
    const float* Wf   = (const float*)d_in[8];
    const float* bf   = (const float*)d_in[9];
    float* out  = (float*)d_out;
    float* sink = (float*)d_ws;   // D_ * F_ floats of scratch

    const size_t shmem = (size_t)LDS_FLOATS * sizeof(float);  // ~192 KB
    chain_kernel<<<D_, 256, shmem, stream>>>(af, pidx, W1, b1, Wm, bm, attw, sink);
    finalize_kernel<<<1, 512, 0, stream>>>(sink, dagw, Wf, bf, out);
}